// Attentive_1580547973848
// MI455X (gfx1250) — compile-verified
//
#include <hip/hip_runtime.h>
#include <stdint.h>

// ---------------------------------------------------------------------------
// out[b,s,:] = x[b,s,:] * w[:]   (x: 16x4096x4096 f32, w: 4096 f32)
// Pure streaming op: ~2.15 GB traffic, 268M FLOP -> 100% HBM-bound on MI455X
// (23.3 TB/s => ~92us floor). Matrix cores are irrelevant at 0.125 FLOP/B;
// the CDNA5 feature that matters is the data-movement path:
//   - gfx1250 async global->LDS copy engine (ASYNCcnt), double-buffered
//   - b128 per-lane traffic, non-temporal store hints (2 GiB streamed once
//     through a 192 MB L2 -> don't write-allocate-thrash it)
//   - wave32-sized blocks (256 thr = 8 waves)
// ---------------------------------------------------------------------------

typedef float v4f __attribute__((ext_vector_type(4)));
// Matches the builtin's parameter type: 'int __attribute__((vector_size(16)))'
typedef int v4i_b128 __attribute__((vector_size(4 * sizeof(int))));

#define BLOCK_SIZE 256   // 8 wave32 waves per workgroup
#define GRID_SIZE  2048  // 524288 threads total; multiple of 1024 (cols4)

#define AS_GLOBAL __attribute__((address_space(1)))
#define AS_LDS    __attribute__((address_space(3)))

#if defined(__HIP_DEVICE_COMPILE__) && defined(__gfx1250__)
#  if __has_builtin(__builtin_amdgcn_global_load_async_to_lds_b128)
#    define USE_ASYNC_COPY 1
#  else
#    define USE_ASYNC_COPY 0
#  endif
#  if __has_builtin(__builtin_amdgcn_s_wait_asynccnt)
#    define HAVE_WAIT_ASYNC_BUILTIN 1
#  else
#    define HAVE_WAIT_ASYNC_BUILTIN 0
#  endif
#else
#  define USE_ASYNC_COPY 0
#  define HAVE_WAIT_ASYNC_BUILTIN 0
#endif

template <int N>
__device__ __forceinline__ void wait_async() {
#if HAVE_WAIT_ASYNC_BUILTIN
  __builtin_amdgcn_s_wait_asynccnt(N);
#else
  asm volatile("s_wait_asynccnt %0" ::"n"(N) : "memory");
#endif
  asm volatile("" ::: "memory");  // keep LDS reads after the wait
}

#if USE_ASYNC_COPY
__device__ __forceinline__ void async_copy_b128(const v4f* gsrc, v4f* lds_dst) {
  __builtin_amdgcn_global_load_async_to_lds_b128(
      (AS_GLOBAL v4i_b128*)gsrc, (AS_LDS v4i_b128*)lds_dst,
      /*offset=*/0, /*cpol=*/0);
}
#endif

__global__ __launch_bounds__(BLOCK_SIZE) void rowscale_f32_kernel(
    const v4f* __restrict__ x4, const v4f* __restrict__ w4,
    v4f* __restrict__ o4, long long n4, int col_mask /* = cols4-1, pow2 */) {
  const long long stride = (long long)gridDim.x * blockDim.x;
  const long long tid0   = (long long)blockIdx.x * blockDim.x + threadIdx.x;

  // Launcher guarantees stride % cols4 == 0 and cols4 is a power of two, so
  // this thread's w-slice is loop-invariant and the column is a single AND.
  const v4f wv = w4[(int)tid0 & col_mask];

#if USE_ASYNC_COPY
  // Double-buffered async global->LDS staging. Each lane owns its private
  // 16B slot, so no workgroup barrier is needed: ASYNCcnt is per-wave and
  // async loads complete in order.
  __shared__ v4f xs[2][BLOCK_SIZE];
  v4f* const slot0 = &xs[0][threadIdx.x];
  v4f* const slot1 = &xs[1][threadIdx.x];

  if (tid0 < n4) {
    async_copy_b128(x4 + tid0, slot0);
  }
  int stage = 0;
  for (long long i = tid0; i < n4; i += stride) {
    const long long inext = i + stride;
    if (inext < n4) {
      async_copy_b128(x4 + inext, stage ? slot0 : slot1);
      wait_async<1>();  // newest may be in flight; oldest (stage) is done
    } else {
      wait_async<0>();  // nothing new issued; drain
    }
    const v4f xv = *(stage ? slot1 : slot0);
    const v4f r  = xv * wv;
    __builtin_nontemporal_store(r, o4 + i);  // NT store: don't thrash L2
    stage ^= 1;
  }
#else
  // Fallback: direct NT b128 loads + software prefetch (global_prefetch_b8).
  for (long long i = tid0; i < n4; i += stride) {
    if (i + stride < n4) __builtin_prefetch(x4 + i + stride, 0, 0);
    const v4f xv = __builtin_nontemporal_load(x4 + i);
    const v4f r  = xv * wv;
    __builtin_nontemporal_store(r, o4 + i);
  }
#endif
}

extern "C" void kernel_launch(void* const* d_in, const int* in_sizes, int n_in,
                              void* d_out, int out_size, void* d_ws, size_t ws_size,
                              hipStream_t stream) {
  (void)n_in; (void)d_ws; (void)ws_size; (void)out_size;
  const v4f* x4 = (const v4f*)d_in[0];
  const v4f* w4 = (const v4f*)d_in[1];
  v4f*       o4 = (v4f*)d_out;

  const long long n     = (long long)in_sizes[0];  // 268,435,456 elements
  const long long n4    = n >> 2;                  // 67,108,864 float4s
  const int       cols4 = in_sizes[1] >> 2;        // 1024 (power of two)

  // GRID_SIZE*BLOCK_SIZE = 524288 threads; 524288 % 1024 == 0, which makes
  // each thread's w float4 loop-invariant (see kernel). ~128 iters/thread.
  rowscale_f32_kernel<<<dim3(GRID_SIZE), dim3(BLOCK_SIZE), 0, stream>>>(
      x4, w4, o4, n4, cols4 - 1);
}